// Voxelizer_62362925137969
// MI455X (gfx1250) — compile-verified
//
#include <hip/hip_runtime.h>

typedef float v2f __attribute__((ext_vector_type(2)));
typedef float v4f __attribute__((ext_vector_type(4)));
typedef float v8f __attribute__((ext_vector_type(8)));

#define ISO_ 0.25f

// ---------------- kernel 1: AABB (per-axis min/max) of verts ----------------
__global__ __launch_bounds__(256) void k_minmax(const float* __restrict__ verts, int nvert,
                                                float* __restrict__ mnmx) {
  __shared__ float smn[3][256];
  __shared__ float smx[3][256];
  int t = threadIdx.x;
  float mnv[3] = {3.0e38f, 3.0e38f, 3.0e38f};
  float mxv[3] = {-3.0e38f, -3.0e38f, -3.0e38f};
  for (int i = t; i < nvert; i += 256) {
#pragma unroll
    for (int d = 0; d < 3; ++d) {
      float v = verts[3 * i + d];
      mnv[d] = fminf(mnv[d], v);
      mxv[d] = fmaxf(mxv[d], v);
    }
  }
#pragma unroll
  for (int d = 0; d < 3; ++d) { smn[d][t] = mnv[d]; smx[d][t] = mxv[d]; }
  __syncthreads();
  for (int s = 128; s > 0; s >>= 1) {
    if (t < s) {
#pragma unroll
      for (int d = 0; d < 3; ++d) {
        smn[d][t] = fminf(smn[d][t], smn[d][t + s]);
        smx[d][t] = fmaxf(smx[d][t], smx[d][t + s]);
      }
    }
    __syncthreads();
  }
  if (t == 0) {
#pragma unroll
    for (int d = 0; d < 3; ++d) { mnmx[d] = smn[d][0]; mnmx[3 + d] = smx[d][0]; }
  }
}

// ---------------- kernel 2: pack verts as B-columns [-2v, |v|^2] ----------------
__global__ __launch_bounds__(256) void k_packverts(const float* __restrict__ verts, int nvert,
                                                   v4f* __restrict__ bv) {
  int i = blockIdx.x * 256 + threadIdx.x;
  if (i >= nvert) return;
  float x = verts[3 * i + 0], y = verts[3 * i + 1], z = verts[3 * i + 2];
  v4f p;
  p.x = -2.0f * x; p.y = -2.0f * y; p.z = -2.0f * z;
  p.w = x * x + y * y + z * z;
  bv[i] = p;
}

// ---------------- kernel 3: per-face constants, packed as four B-columns ----------------
// All four ray-cast conditions become "WMMA result >= 0":
//   nq-col = [-nrm*denom, +v0n*denom]          ->  D = -q*denom  ( >=0  <=> t >= 0 )
//   a-col  = [dir x e2, -u_c]                  ->  D = a         ( >=0  <=> u >= 0 )
//   b-col  = [e1 x dir, -v_c]                  ->  D = b         ( >=0  <=> v >= 0 )
//   t-col  = [-(dxe2 + e1xd), nn + u_c + v_c]  ->  D = nn-a-b    ( >=0  <=> u+v <= 1 )
// Degenerate faces (denom==0) are killed by forcing t-col constant to -3e38.
__global__ __launch_bounds__(256) void k_packfaces(const float* __restrict__ verts,
                                                   const int* __restrict__ faces, int nface,
                                                   v4f* __restrict__ nq4, v4f* __restrict__ a4,
                                                   v4f* __restrict__ b4, v4f* __restrict__ t4) {
  int f = blockIdx.x * 256 + threadIdx.x;
  if (f >= nface) return;
  int i0 = faces[3 * f + 0], i1 = faces[3 * f + 1], i2 = faces[3 * f + 2];
  float v0x = verts[3 * i0 + 0], v0y = verts[3 * i0 + 1], v0z = verts[3 * i0 + 2];
  float v1x = verts[3 * i1 + 0], v1y = verts[3 * i1 + 1], v1z = verts[3 * i1 + 2];
  float v2x = verts[3 * i2 + 0], v2y = verts[3 * i2 + 1], v2z = verts[3 * i2 + 2];
  float e1x = v1x - v0x, e1y = v1y - v0y, e1z = v1z - v0z;
  float e2x = v2x - v0x, e2y = v2y - v0y, e2z = v2z - v0z;
  // nrm = e1 x e2
  float nx = e1y * e2z - e1z * e2y;
  float ny = e1z * e2x - e1x * e2z;
  float nz = e1x * e2y - e1y * e2x;
  float nn = nx * nx + ny * ny + nz * nz;
  float v0n = v0x * nx + v0y * ny + v0z * nz;
  float denom = nx;  // dir=(1,0,0) => denom = nrm.x
  // dir x e2 = (0, -e2z, e2y) ;  e1 x dir = (0, e1z, -e1y)
  float ay = -e2z, az = e2y;
  float by = e1z,  bz = -e1y;
  float u_c = ay * v0y + az * v0z;
  float v_c = by * v0y + bz * v0z;

  v4f nq; nq.x = -nx * denom; nq.y = -ny * denom; nq.z = -nz * denom; nq.w = v0n * denom;
  v4f a;  a.x = 0.0f; a.y = ay; a.z = az; a.w = -u_c;
  v4f b;  b.x = 0.0f; b.y = by; b.z = bz; b.w = -v_c;
  v4f t;  t.x = 0.0f; t.y = -(ay + by); t.z = -(az + bz);
  t.w = (denom != 0.0f) ? (nn + u_c + v_c) : -3.0e38f;

  nq4[f] = nq; a4[f] = a; b4[f] = b; t4[f] = t;
}

// ---------------- kernel 4: WMMA voxelizer ----------------
// One wave per 16-grid-point tile.  A (16x4 f32) = [cx, cy, cz, 1] rows.
// A-matrix VGPR layout: lanes 0-15 hold {K0,K1}, lanes 16-31 hold {K2,K3} (M = lane&15).
// B-matrix VGPR layout (mirror): lanes 0-15 hold rows {K0,K1}, lanes 16-31 rows {K2,K3}
//   for column N = lane&15.
// D layout: VGPR r, lane l -> row M = 8*(l>=16)+r, col N = l&15.
__global__ __launch_bounds__(256) void k_voxel(const float* __restrict__ mnmx,
                                               const v4f* __restrict__ bv, int nvert,
                                               const v4f* __restrict__ nq4,
                                               const v4f* __restrict__ a4,
                                               const v4f* __restrict__ b4,
                                               const v4f* __restrict__ t4, int nface,
                                               float* __restrict__ out, int npts) {
  const int l = threadIdx.x & 31;
  const int wave = blockIdx.x * (blockDim.x >> 5) + (threadIdx.x >> 5);
  if (wave * 16 >= npts) return;          // wave-uniform: EXEC stays all-ones
  const int hl = l >> 4;                  // which 16-lane group
  const int m = l & 15;                   // my A-row / B-column within the tile

  // grid point coordinates for M = m (flat index g = ix*1024 + iy*32 + iz)
  const int g = wave * 16 + m;
  const int ix = g >> 10, iy = (g >> 5) & 31, iz = g & 31;
  const float mnx = mnmx[0], mny = mnmx[1], mnz = mnmx[2];
  const float sx = mnmx[3] - mnx, sy = mnmx[4] - mny, sz = mnmx[5] - mnz;
  const float inv31 = 1.0f / 31.0f;
  const float cx = mnx + ((float)ix * inv31) * sx;
  const float cy = mny + ((float)iy * inv31) * sy;
  const float cz = mnz + ((float)iz * inv31) * sz;
  const float c2 = cx * cx + cy * cy + cz * cz;

  v2f A;
  A.x = hl ? cz : cx;
  A.y = hl ? 1.0f : cy;
  const v8f zc = {0.f, 0.f, 0.f, 0.f, 0.f, 0.f, 0.f, 0.f};

  // -------- phase 1: min over verts of (|v|^2 - 2 c.v) --------
  v8f accw;
#pragma unroll
  for (int r = 0; r < 8; ++r) accw[r] = 3.0e38f;

  const int nvt = nvert >> 4;
  for (int n = 0; n < nvt; ++n) {
    __builtin_prefetch(&bv[(n + 8) * 16 + m], 0, 3);
    v4f P = bv[n * 16 + m];
    v2f B;
    B.x = hl ? P.z : P.x;
    B.y = hl ? P.w : P.y;
    v8f D = __builtin_amdgcn_wmma_f32_16x16x4_f32(false, A, false, B, (short)0, zc, false, false);
#pragma unroll
    for (int r = 0; r < 8; ++r) accw[r] = fminf(accw[r], D[r]);
  }

  // -------- phase 2: ray-cast parity over faces (4 WMMAs, hit = min4 >= 0) --------
  unsigned par = 0;
  const int nft = nface >> 4;
  for (int n = 0; n < nft; ++n) {
    const int fi = n * 16 + m;
    __builtin_prefetch(&nq4[fi + 128], 0, 3);
    __builtin_prefetch(&a4[fi + 128], 0, 3);
    __builtin_prefetch(&b4[fi + 128], 0, 3);
    __builtin_prefetch(&t4[fi + 128], 0, 3);
    v4f Pq = nq4[fi];
    v4f Pa = a4[fi];
    v4f Pb = b4[fi];
    v4f Pt = t4[fi];
    v2f Bq, Ba, Bb, Bt;
    Bq.x = hl ? Pq.z : Pq.x; Bq.y = hl ? Pq.w : Pq.y;
    Ba.x = hl ? Pa.z : Pa.x; Ba.y = hl ? Pa.w : Pa.y;
    Bb.x = hl ? Pb.z : Pb.x; Bb.y = hl ? Pb.w : Pb.y;
    Bt.x = hl ? Pt.z : Pt.x; Bt.y = hl ? Pt.w : Pt.y;
    v8f Dq = __builtin_amdgcn_wmma_f32_16x16x4_f32(false, A, false, Bq, (short)0, zc, false, false);
    v8f Da = __builtin_amdgcn_wmma_f32_16x16x4_f32(false, A, false, Ba, (short)0, zc, false, false);
    v8f Db = __builtin_amdgcn_wmma_f32_16x16x4_f32(false, A, false, Bb, (short)0, zc, false, false);
    v8f Dt = __builtin_amdgcn_wmma_f32_16x16x4_f32(false, A, false, Bt, (short)0, zc, false, false);
#pragma unroll
    for (int r = 0; r < 8; ++r) {
      float mm = fminf(fminf(Dq[r], Da[r]), fminf(Db[r], Dt[r]));
      par ^= (mm >= 0.0f) ? (1u << r) : 0u;
    }
  }

  // -------- reductions over the 16 columns (within each 16-lane group) --------
#pragma unroll
  for (int r = 0; r < 8; ++r) {
    float v = accw[r];
    v = fminf(v, __shfl_xor(v, 1, 32));
    v = fminf(v, __shfl_xor(v, 2, 32));
    v = fminf(v, __shfl_xor(v, 4, 32));
    v = fminf(v, __shfl_xor(v, 8, 32));
    accw[r] = v;
  }
  par ^= __shfl_xor(par, 1, 32);
  par ^= __shfl_xor(par, 2, 32);
  par ^= __shfl_xor(par, 4, 32);
  par ^= __shfl_xor(par, 8, 32);

  // -------- finalize: sdf = inside ? -dist : dist ; clamp band --------
  float vals[8];
#pragma unroll
  for (int r = 0; r < 8; ++r) {
    const int M = hl * 8 + r;                 // my output row
    const float c2M = __shfl(c2, M, 32);      // |c|^2 lives in lane M (computed for m==M)
    const float dist = sqrtf(fmaxf(accw[r] + c2M, 0.0f));
    const float sdf = ((par >> r) & 1u) ? -dist : dist;
    vals[r] = (fabsf(sdf) <= ISO_) ? sdf : 0.0f;
  }
  if (m == 0) {                               // lanes 0 and 16 write 8 results each
#pragma unroll
    for (int r = 0; r < 8; ++r) out[wave * 16 + hl * 8 + r] = vals[r];
  }
}

extern "C" void kernel_launch(void* const* d_in, const int* in_sizes, int n_in,
                              void* d_out, int out_size, void* d_ws, size_t ws_size,
                              hipStream_t stream) {
  const float* verts = (const float*)d_in[0];
  const int* faces = (const int*)d_in[1];
  float* out = (float*)d_out;
  const int nvert = in_sizes[0] / 3;   // 8192
  const int nface = in_sizes[1] / 3;   // 16384
  const int npts = out_size;           // 32768

  char* ws = (char*)d_ws;
  size_t off = 0;
  float* mnmx = (float*)(ws + off);            off += 256;
  v4f* bv = (v4f*)(ws + off);                  off += (size_t)nvert * sizeof(v4f);
  off = (off + 255) & ~(size_t)255;
  v4f* nq4 = (v4f*)(ws + off);                 off += (size_t)nface * sizeof(v4f);
  v4f* a4 = (v4f*)(ws + off);                  off += (size_t)nface * sizeof(v4f);
  v4f* b4 = (v4f*)(ws + off);                  off += (size_t)nface * sizeof(v4f);
  v4f* t4 = (v4f*)(ws + off);                  off += (size_t)nface * sizeof(v4f);
  (void)ws_size; (void)n_in;

  k_minmax<<<1, 256, 0, stream>>>(verts, nvert, mnmx);
  k_packverts<<<(nvert + 255) / 256, 256, 0, stream>>>(verts, nvert, bv);
  k_packfaces<<<(nface + 255) / 256, 256, 0, stream>>>(verts, faces, nface, nq4, a4, b4, t4);

  const int waves = npts / 16;                  // 2048 M-tiles
  const int blocks = (waves + 7) / 8;           // 8 waves (256 threads) per block
  k_voxel<<<blocks, 256, 0, stream>>>(mnmx, bv, nvert, nq4, a4, b4, t4, nface, out, npts);
}